// SelectiveSSM_5566277616112
// MI455X (gfx1250) — compile-verified
//
#include <hip/hip_runtime.h>
#include <math.h>

typedef __attribute__((ext_vector_type(2))) float v2f;
typedef __attribute__((ext_vector_type(8))) float v8f;

#define B_SZ    2
#define T_LEN   2048
#define DMODEL  512
#define DINNER  1024
#define DSTATE  16
#define BT      (B_SZ * T_LEN)      // 4096
#define XPW     (2 * DSTATE + 1)    // 33

// GEMM tiling
#define BM  128
#define BN  64
#define BK  32
#define BKP 36        // padded LDS row stride (dwords): 16B-aligned, conflict-free

// ---------------------------------------------------------------------------
// CDNA5 async global->LDS staging (ASYNCcnt path), with safe fallback.
// ---------------------------------------------------------------------------
#define USE_ASYNC_LDS 1

__device__ __forceinline__ void copy_b128_to_lds(float* l, const float* g) {
#if USE_ASYNC_LDS
    // Per-lane: copies 16 bytes from the global address in vaddr to the LDS address in vdst, tracked by ASYNCcnt.
    asm volatile("global_load_async_to_lds_b128 %0, %1, off"
                 :: "v"((unsigned)(uintptr_t)l), "v"(g)
                 : "memory");
#else
    *(float4*)l = *(const float4*)g;
#endif
}

__device__ __forceinline__ void wait_async_lds() {
#if USE_ASYNC_LDS
#if __has_builtin(__builtin_amdgcn_s_wait_asynccnt)
    __builtin_amdgcn_s_wait_asynccnt(0);
#else
    asm volatile("s_wait_asynccnt 0x0" ::: "memory");
#endif
#endif
}

// ---------------------------------------------------------------------------
// fp32 WMMA GEMM:  C[M,N] = A[M,K] * Bw[N,K]^T   (row-major, K % 32 == 0)
// Block = 256 thr (8 waves). Block tile 128x64, K-block 32, double-buffered
// LDS staging via async-to-LDS; wave computes a 16x64 strip (4 v8f accums).
// ---------------------------------------------------------------------------
__global__ __launch_bounds__(256)
void gemm_nt_wmma(const float* __restrict__ A, const float* __restrict__ Bw,
                  float* __restrict__ C, int M, int N, int K) {
    __shared__ float As[2][BM * BKP];
    __shared__ float Bs[2][BN * BKP];

    const int tid  = threadIdx.x;
    const int lane = tid & 31;
    const int wid  = tid >> 5;
    const int half = lane >> 4;      // 0: K pair {0,1}, 1: K pair {2,3}
    const int l    = lane & 15;

    const int m0 = blockIdx.y * BM;
    const int n0 = blockIdx.x * BN;
    const int aRow = wid * 16 + l;   // this wave's fragment row within A tile

    v8f acc0 = {}; v8f acc1 = {}; v8f acc2 = {}; v8f acc3 = {};

    const float* gA = A  + (size_t)m0 * K;
    const float* gB = Bw + (size_t)n0 * K;

    // Cooperative tile loader: 16B per lane, 8 lanes cover one 128B row chunk.
    auto load_tile = [&](int p, int kb) {
        const float* ga = gA + kb * BK;
        #pragma unroll
        for (int r = 0; r < 4; ++r) {              // A: 1024 float4
            int i   = tid + 256 * r;
            int row = i >> 3;
            int kq  = (i & 7) * 4;
            copy_b128_to_lds(&As[p][row * BKP + kq], ga + (size_t)row * K + kq);
        }
        const float* gb = gB + kb * BK;
        #pragma unroll
        for (int r = 0; r < 2; ++r) {              // B: 512 float4
            int i   = tid + 256 * r;
            int row = i >> 3;
            int kq  = (i & 7) * 4;
            copy_b128_to_lds(&Bs[p][row * BKP + kq], gb + (size_t)row * K + kq);
        }
    };

    const int nkb = K / BK;
    load_tile(0, 0);
    wait_async_lds();
    __syncthreads();

    int p = 0;
    for (int kb = 0; kb < nkb; ++kb) {
        if (kb + 1 < nkb) load_tile(p ^ 1, kb + 1);   // prefetch next K-block

        // Compute from LDS buffer p: 8 iterations x 4 WMMA.
        #pragma unroll
        for (int kk = 0; kk < BK; kk += 4) {
            int ko = kk + 2 * half;
            v2f a  = *(const v2f*)&As[p][aRow * BKP + ko];
            v2f b0 = *(const v2f*)&Bs[p][( 0 + l) * BKP + ko];
            v2f b1 = *(const v2f*)&Bs[p][(16 + l) * BKP + ko];
            v2f b2 = *(const v2f*)&Bs[p][(32 + l) * BKP + ko];
            v2f b3 = *(const v2f*)&Bs[p][(48 + l) * BKP + ko];
            acc0 = __builtin_amdgcn_wmma_f32_16x16x4_f32(false, a, false, b0, (short)0, acc0, false, false);
            acc1 = __builtin_amdgcn_wmma_f32_16x16x4_f32(false, a, false, b1, (short)0, acc1, false, false);
            acc2 = __builtin_amdgcn_wmma_f32_16x16x4_f32(false, a, false, b2, (short)0, acc2, false, false);
            acc3 = __builtin_amdgcn_wmma_f32_16x16x4_f32(false, a, false, b3, (short)0, acc3, false, false);
        }

        wait_async_lds();
        __syncthreads();
        p ^= 1;
    }

    // D layout: VGPR r -> rows m0w+r (lanes 0-15) and m0w+8+r (lanes 16-31)
    const int mBase = m0 + wid * 16;
    #pragma unroll
    for (int r = 0; r < 8; ++r) {
        int row = mBase + r + 8 * half;
        float* crow = C + (size_t)row * N + n0 + l;
        crow[ 0] = acc0[r];
        crow[16] = acc1[r];
        crow[32] = acc2[r];
        crow[48] = acc3[r];
    }
}

// ---------------------------------------------------------------------------
// Depthwise causal conv (k=4) + bias + SiLU.  x_ssm = xz[..., :DINNER].
// ---------------------------------------------------------------------------
__global__ __launch_bounds__(256)
void conv_silu_kernel(const float* __restrict__ xz, const float* __restrict__ conv_w,
                      const float* __restrict__ conv_b, float* __restrict__ x_conv) {
    int idx = blockIdx.x * 256 + threadIdx.x;            // (b*T + t)*DINNER + d
    if (idx >= BT * DINNER) return;
    int d  = idx & (DINNER - 1);
    int bt = idx >> 10;
    int t  = bt & (T_LEN - 1);

    float w0 = conv_w[d * 4 + 0];
    float w1 = conv_w[d * 4 + 1];
    float w2 = conv_w[d * 4 + 2];
    float w3 = conv_w[d * 4 + 3];

    const float* xs = xz + (size_t)bt * (2 * DINNER) + d;
    float s = conv_b[d] + w3 * xs[0];
    if (t >= 1) s += w2 * xs[-(ptrdiff_t)(1 * 2 * DINNER)];
    if (t >= 2) s += w1 * xs[-(ptrdiff_t)(2 * 2 * DINNER)];
    if (t >= 3) s += w0 * xs[-(ptrdiff_t)(3 * 2 * DINNER)];

    x_conv[idx] = s / (1.0f + expf(-s));                 // SiLU
}

// ---------------------------------------------------------------------------
// xp[b,t,j] = sum_d x_conv[b,t,d] * W_x[j,d] ;   j in [0,33)
// ---------------------------------------------------------------------------
__global__ __launch_bounds__(256)
void xp_kernel(const float* __restrict__ x_conv, const float* __restrict__ W_x,
               float* __restrict__ xp) {
    int gw   = (blockIdx.x * 256 + threadIdx.x) >> 5;    // global wave index
    int lane = threadIdx.x & 31;
    if (gw >= BT * XPW) return;
    int j  = gw % XPW;
    int bt = gw / XPW;

    const float* xr = x_conv + (size_t)bt * DINNER;
    const float* wr = W_x    + (size_t)j  * DINNER;
    float s = 0.0f;
    for (int d = lane; d < DINNER; d += 32) s = fmaf(xr[d], wr[d], s);
    for (int off = 16; off; off >>= 1) s += __shfl_xor(s, off, 32);
    if (lane == 0) xp[(size_t)bt * XPW + j] = s;
}

// ---------------------------------------------------------------------------
// delta = clip(softplus(dt_raw*W_dt+b_dt) * (1+tanh(reg*W_reg+b_reg)), 1e-6, 5)
// ---------------------------------------------------------------------------
__global__ __launch_bounds__(256)
void delta_kernel(const float* __restrict__ xp, const float* __restrict__ regime,
                  const float* __restrict__ W_dt, const float* __restrict__ b_dt,
                  const float* __restrict__ W_reg, const float* __restrict__ b_reg,
                  float* __restrict__ delta) {
    int idx = blockIdx.x * 256 + threadIdx.x;
    if (idx >= BT * DINNER) return;
    int d  = idx & (DINNER - 1);
    int bt = idx >> 10;

    float dt_raw = xp[(size_t)bt * XPW + 32];
    float raw = fmaf(dt_raw, W_dt[d], b_dt[d]);
    float sp  = (raw > 20.0f) ? raw : log1pf(expf(raw));        // softplus
    float g   = 1.0f + tanhf(fmaf(regime[bt], W_reg[d], b_reg[d]));
    float dl  = fminf(fmaxf(sp * g, 1e-6f), 5.0f);
    delta[idx] = dl;
}

// ---------------------------------------------------------------------------
// Selective scan. Lane = (b,d,n); 16 lanes per (b,d) group; state in regs.
// Running clamp of cumsum is valid (dA < 0 always => monotone decreasing).
// Fused with D-skip and z-gating (SiLU).
// ---------------------------------------------------------------------------
__global__ __launch_bounds__(256)
void scan_kernel(const float* __restrict__ delta, const float* __restrict__ x_conv,
                 const float* __restrict__ xp, const float* __restrict__ xz,
                 const float* __restrict__ A_log, const float* __restrict__ Dv,
                 float* __restrict__ y_gated) {
    int gid = blockIdx.x * 256 + threadIdx.x;
    int n   = gid & 15;
    int grp = gid >> 4;                     // b*DINNER + d
    if (grp >= B_SZ * DINNER) return;
    int d = grp & (DINNER - 1);
    int b = grp >> 10;

    const float An = -expf(A_log[d * DSTATE + n]);
    const float Dd = Dv[d];

    float S = 0.0f, acc = 0.0f;
    const size_t base = (size_t)b * T_LEN;

    for (int t = 0; t < T_LEN; ++t) {
        const size_t bt = base + t;
        float dl = delta [bt * DINNER + d];
        float xv = x_conv[bt * DINNER + d];
        float Bn = xp[bt * XPW + n];
        float Cn = xp[bt * XPW + DSTATE + n];

        S = fminf(fmaxf(fmaf(dl, An, S), -15.0f), 15.0f);
        float eS = expf(S);
        acc = fmaf(dl * Bn * xv, expf(-S), acc);
        float contrib = eS * acc * Cn;

        contrib += __shfl_xor(contrib, 8, 32);
        contrib += __shfl_xor(contrib, 4, 32);
        contrib += __shfl_xor(contrib, 2, 32);
        contrib += __shfl_xor(contrib, 1, 32);

        if (n == 0) {
            float zv = xz[bt * (2 * DINNER) + DINNER + d];
            float y  = fmaf(Dd, xv, contrib);
            y *= zv / (1.0f + expf(-zv));   // * silu(z)
            y_gated[bt * DINNER + d] = y;
        }
    }
}

// ---------------------------------------------------------------------------
extern "C" void kernel_launch(void* const* d_in, const int* in_sizes, int n_in,
                              void* d_out, int out_size, void* d_ws, size_t ws_size,
                              hipStream_t stream) {
    const float* x       = (const float*)d_in[0];
    const float* regime  = (const float*)d_in[1];
    const float* W_in    = (const float*)d_in[2];
    const float* conv_w  = (const float*)d_in[3];
    const float* conv_b  = (const float*)d_in[4];
    const float* A_log   = (const float*)d_in[5];
    const float* Dv      = (const float*)d_in[6];
    const float* W_x     = (const float*)d_in[7];
    const float* W_dt    = (const float*)d_in[8];
    const float* b_dt    = (const float*)d_in[9];
    const float* W_reg   = (const float*)d_in[10];
    const float* b_reg   = (const float*)d_in[11];
    const float* W_out   = (const float*)d_in[12];
    float* out = (float*)d_out;

    float* ws      = (float*)d_ws;
    float* xz      = ws;                                  // BT * 2048
    float* x_conv  = xz     + (size_t)BT * 2 * DINNER;    // BT * 1024
    float* xp      = x_conv + (size_t)BT * DINNER;        // BT * 33
    float* delta   = xp     + (size_t)BT * XPW;           // BT * 1024
    float* y_gated = delta  + (size_t)BT * DINNER;        // BT * 1024

    dim3 blk(256);

    // 1) xz = x @ W_in^T : M=4096, N=2048, K=512
    gemm_nt_wmma<<<dim3((2 * DINNER) / BN, BT / BM), blk, 0, stream>>>(
        x, W_in, xz, BT, 2 * DINNER, DMODEL);

    // 2) depthwise conv + SiLU
    conv_silu_kernel<<<(BT * DINNER + 255) / 256, blk, 0, stream>>>(
        xz, conv_w, conv_b, x_conv);

    // 3) xp = x_conv @ W_x^T  (33 outputs per (b,t))
    xp_kernel<<<(BT * XPW * 32 + 255) / 256, blk, 0, stream>>>(
        x_conv, W_x, xp);

    // 4) delta
    delta_kernel<<<(BT * DINNER + 255) / 256, blk, 0, stream>>>(
        xp, regime, W_dt, b_dt, W_reg, b_reg, delta);

    // 5) selective scan + D-skip + z gate
    scan_kernel<<<(B_SZ * DINNER * 16 + 255) / 256, blk, 0, stream>>>(
        delta, x_conv, xp, xz, A_log, Dv, y_gated);

    // 6) out = y_gated @ W_out^T : M=4096, N=512, K=1024
    gemm_nt_wmma<<<dim3(DMODEL / BN, BT / BM), blk, 0, stream>>>(
        y_gated, W_out, out, BT, DMODEL, DINNER);
}